// ContactDiffusion_7413113553496
// MI455X (gfx1250) — compile-verified
//
#include <hip/hip_runtime.h>

typedef float v2f __attribute__((ext_vector_type(2)));
typedef float v4f __attribute__((ext_vector_type(4)));
typedef float v8f __attribute__((ext_vector_type(8)));

#define NN       8192
#define DLAT     64
#define ROWS     16       // rows of K per block
#define CHUNK    64       // columns per phase-B chunk
#define THREADS  128      // 4 waves (wave32)
#define LSTRIDE  68       // padded LDS row stride for K tile (floats)
#define LATSTR   80       // padded LDS row stride for latent tile (floats): 80%64=16 -> conflict-free

// k_ij = (D+1e-8)^(-0.5*(ai+aj)) * exp(-D/12)
//      = exp2( -(0.5*ai+0.5*aj)*log2(D+1e-8) - D*log2(e)/12 )
__device__ __forceinline__ float kfun(float d, float half_ai, float aj, bool diag) {
    const float C12 = 0.120112240878644979f;            // log2(e)/12
    float a = half_ai + 0.5f * aj;
    float t = __builtin_amdgcn_logf(d + 1e-8f);         // v_log_f32 (log2)
    float e = -(a * t + d * C12);
    float k = __builtin_amdgcn_exp2f(e);                // v_exp_f32 (exp2)
    return diag ? 0.0f : k;
}

// Async-stage latent[j0 .. j0+63][0..63] into padded LDS tile (16 KB -> 40 KB padded x2 buf).
// Each of 128 threads issues 8x b128 (one float4 per lane per issue): 1024 slots total.
__device__ __forceinline__ void issue_lat_async(const float* __restrict__ latent,
                                                int j0, float* s_buf, int tid) {
#pragma unroll
    for (int it = 0; it < 8; ++it) {
        const int s  = tid + THREADS * it;       // 0..1023
        const int r  = s >> 4;                   // latent row within chunk (0..63)
        const int c4 = (s & 15) << 2;            // col (0..60 step 4)
        const float* g = latent + (long)(j0 + r) * DLAT + c4;
        const unsigned laddr = (unsigned)(uintptr_t)(s_buf + r * LATSTR + c4);
        asm volatile("global_load_async_to_lds_b128 %0, %1, off"
                     :: "v"(laddr), "v"(g) : "memory");
    }
}

__global__ __launch_bounds__(THREADS)
void contact_diffusion_kernel(const float* __restrict__ latent,
                              const float* __restrict__ D,
                              const float* __restrict__ alpha,
                              float* __restrict__ out_mm,
                              float* __restrict__ out_K)
{
    __shared__ float s_tile[ROWS * LSTRIDE];        // normalized K chunk (16 x 64, padded)
    __shared__ float s_lat[2][CHUNK * LATSTR];      // double-buffered latent slab (64 x 64, padded)
    __shared__ float s_part[THREADS];
    __shared__ float s_inv[ROWS];

    const int tid  = threadIdx.x;
    const int lane = tid & 31;
    const int wave = tid >> 5;                 // 0..3 -> d-tile
    const int i0   = blockIdx.x * ROWS;

    const int row  = tid >> 3;                 // 0..15 (8 threads per row)
    const int cs   = (tid & 7) * 8;            // 8 consecutive cols per thread
    const float half_ai = 0.5f * alpha[i0 + row];
    const long  rowD    = (long)(i0 + row) * NN;

    // ---------------- Phase A: row sums (no stores) ----------------
    float psum = 0.f;
    for (int j0 = 0; j0 < NN; j0 += CHUNK) {
        const int jb = j0 + cs;
        v4f d0 = *(const v4f*)(D + rowD + jb);
        v4f d1 = *(const v4f*)(D + rowD + jb + 4);
        v4f a0 = *(const v4f*)(alpha + jb);
        v4f a1 = *(const v4f*)(alpha + jb + 4);
#pragma unroll
        for (int u = 0; u < 4; ++u) {
            psum += kfun(d0[u], half_ai, a0[u], (jb + u)     == (i0 + row));
            psum += kfun(d1[u], half_ai, a1[u], (jb + 4 + u) == (i0 + row));
        }
    }
    s_part[tid] = psum;
    __syncthreads();
    if (tid < ROWS) {
        float s = 0.f;
#pragma unroll
        for (int u = 0; u < 8; ++u) s += s_part[tid * 8 + u];
        s_inv[tid] = 1.0f / (s + 1e-8f);
    }
    __syncthreads();
    const float inv = s_inv[row];

    // ---------------- Phase B: normalized K -> global + LDS, async latent, WMMA ----------------
    v8f c = {0.f, 0.f, 0.f, 0.f, 0.f, 0.f, 0.f, 0.f};
    const int hi   = lane >> 4;                // 0/1
    const int lmod = lane & 15;
    const float* abase = s_tile + lmod * LSTRIDE + 2 * hi;      // A[M=lmod][K=kk*4+2*hi+v]

    issue_lat_async(latent, 0, s_lat[0], tid);                  // prefetch first latent slab

    for (int j0 = 0; j0 < NN; j0 += CHUNK) {
        const int buf = (j0 / CHUNK) & 1;
        const int jb  = j0 + cs;
        __builtin_prefetch(D + rowD + jb + CHUNK, 0, 0);        // global_prefetch_b8
        v4f d0 = *(const v4f*)(D + rowD + jb);
        v4f d1 = *(const v4f*)(D + rowD + jb + 4);
        v4f a0 = *(const v4f*)(alpha + jb);
        v4f a1 = *(const v4f*)(alpha + jb + 4);
        v4f k0, k1;
#pragma unroll
        for (int u = 0; u < 4; ++u) {
            k0[u] = inv * kfun(d0[u], half_ai, a0[u], (jb + u)     == (i0 + row));
            k1[u] = inv * kfun(d1[u], half_ai, a1[u], (jb + 4 + u) == (i0 + row));
        }
        // normalized K tile out (coalesced b128 x2 per thread)
        *(v4f*)(out_K + rowD + jb)     = k0;
        *(v4f*)(out_K + rowD + jb + 4) = k1;
        // LDS tile for WMMA A operand
        float* lp = s_tile + row * LSTRIDE + cs;
        *(v4f*)(lp)     = k0;
        *(v4f*)(lp + 4) = k1;

        // latent slab for this chunk must have landed; K tile must be visible
        asm volatile("s_wait_asynccnt 0x0" ::: "memory");
        __syncthreads();

        // kick off next slab into the other buffer (overlaps with WMMA + next compute)
        if (j0 + CHUNK < NN)
            issue_lat_async(latent, j0 + CHUNK, s_lat[buf ^ 1], tid);

        // each wave: 16 x V_WMMA_F32_16X16X4_F32 over this 64-wide K slab
        // B[K][N]: K = kk*4 + 2*hi + v, N = wave*16 + lmod  (from padded LDS slab)
        const float* lb = s_lat[buf] + (2 * hi) * LATSTR + wave * 16 + lmod;
#pragma unroll
        for (int kk = 0; kk < CHUNK / 4; ++kk) {
            v2f A, B;
            A.x = abase[kk * 4 + 0];                 // ds_load_b64 (consecutive)
            A.y = abase[kk * 4 + 1];
            B.x = lb[(kk * 4) * LATSTR];             // ds_load_2addr_b32 (offset1:80)
            B.y = lb[(kk * 4 + 1) * LATSTR];
            c = __builtin_amdgcn_wmma_f32_16x16x4_f32(
                    /*neg_a=*/false, A, /*neg_b=*/false, B,
                    /*c_mod=*/(short)0, c, /*reuse_a=*/false, /*reuse_b=*/false);
        }
        __syncthreads();
    }

    // C/D layout: VGPR r, lane L -> M = r + 8*(L/16), N = L%16
    float* obase = out_mm + (long)(i0 + 8 * hi) * DLAT + wave * 16 + lmod;
#pragma unroll
    for (int r = 0; r < 8; ++r) obase[r * DLAT] = c[r];
}

extern "C" void kernel_launch(void* const* d_in, const int* in_sizes, int n_in,
                              void* d_out, int out_size, void* d_ws, size_t ws_size,
                              hipStream_t stream) {
    const float* latent = (const float*)d_in[0];   // 8192 x 64
    const float* D      = (const float*)d_in[1];   // 8192 x 8192
    const float* alpha  = (const float*)d_in[2];   // 8192
    float* out_mm = (float*)d_out;                 // 8192 x 64   (first return)
    float* out_K  = out_mm + (long)NN * DLAT;      // 8192 x 8192 (second return)
    (void)in_sizes; (void)n_in; (void)out_size; (void)d_ws; (void)ws_size;

    contact_diffusion_kernel<<<dim3(NN / ROWS), dim3(THREADS), 0, stream>>>(
        latent, D, alpha, out_mm, out_K);
}